// LengthRegulator_46445776339408
// MI455X (gfx1250) — compile-verified
//
#include <hip/hip_runtime.h>
#include <stdint.h>

// Problem constants from the reference (static shapes).
#define BB    16
#define TT    512
#define DD    512
#define TOUT  2560

typedef float v4f __attribute__((ext_vector_type(4)));
typedef int   v4i __attribute__((vector_size(16)));   // matches builtin's V4i
#define AS1 __attribute__((address_space(1)))
#define AS3 __attribute__((address_space(3)))

// ---------------------------------------------------------------------------
// Kernel 1: per-batch inclusive scan of rounded durations -> ends[B][T]
// ---------------------------------------------------------------------------
__global__ __launch_bounds__(TT) void lr_scan_kernel(const float* __restrict__ dur,
                                                     int* __restrict__ ends) {
  __shared__ int s[TT];
  const int b = blockIdx.x;
  const int t = threadIdx.x;
  // (dur + 0.5).floor() in float32 — identical arithmetic to the reference.
  s[t] = (int)floorf(dur[b * TT + t] + 0.5f);
  __syncthreads();
#pragma unroll
  for (int off = 1; off < TT; off <<= 1) {
    int v = (t >= off) ? s[t - off] : 0;
    __syncthreads();
    s[t] += v;
    __syncthreads();
  }
  ends[b * TT + t] = s[t];
}

// ---------------------------------------------------------------------------
// Kernel 2: expand. One wave32 per output frame; 8 frames per 256-thread block.
// The 2KB ends row is staged into LDS with gfx1250 async global->LDS copies.
// ---------------------------------------------------------------------------
__global__ __launch_bounds__(256) void lr_expand_kernel(const float* __restrict__ x,
                                                        const int* __restrict__ ends,
                                                        float* __restrict__ out) {
  __shared__ int s_ends[TT];
  const int tid = threadIdx.x;
  const int blocks_per_batch = TOUT / 8;  // 320
  const int b    = blockIdx.x / blocks_per_batch;
  const int row0 = (blockIdx.x % blocks_per_batch) * 8;

  const int* erow = ends + b * TT;

#if __has_builtin(__builtin_amdgcn_global_load_async_to_lds_b128)
  // CDNA5 async path: 128 lanes each move 16B global -> LDS,
  // tracked on ASYNCcnt (no VGPR round trip).
  if (tid < TT / 4) {
    AS1 v4i* g = (AS1 v4i*)(uintptr_t)(erow + tid * 4);
    AS3 v4i* l = (AS3 v4i*)(uint32_t)(uintptr_t)(&s_ends[tid * 4]);
    __builtin_amdgcn_global_load_async_to_lds_b128(g, l, /*offset=*/0, /*cpol=*/0);
  }
#if __has_builtin(__builtin_amdgcn_s_wait_asynccnt)
  __builtin_amdgcn_s_wait_asynccnt(0);
#else
  asm volatile("s_wait_asynccnt 0" ::: "memory");
#endif
#else
  // Fallback: plain staged copy.
  for (int i = tid; i < TT; i += 256) s_ends[i] = erow[i];
#endif
  __syncthreads();

  const int wave = tid >> 5;
  const int lane = tid & 31;
  const int j = row0 + wave;              // output frame owned by this wave
  const int total = s_ends[TT - 1];

  // first t with ends[t] > j  (== searchsorted(ends, j, side='right'))
  int lo = 0, hi = TT;
  while (lo < hi) {                        // at most 10 wave-uniform iterations, LDS broadcast reads
    int mid = (lo + hi) >> 1;
    if (s_ends[mid] > j) hi = mid; else lo = mid + 1;
  }
  const int src = (lo < TT) ? lo : (TT - 1);

  v4f* __restrict__ op = (v4f*)(out + ((size_t)b * TOUT + j) * DD);
  if (j < total) {
    const v4f* __restrict__ xs = (const v4f*)(x + ((size_t)b * TT + src) * DD);
#pragma unroll
    for (int k = 0; k < 4; ++k) {
      v4f val = xs[lane + k * 32];                          // global_load_b128 (cached, reused)
      __builtin_nontemporal_store(val, &op[lane + k * 32]); // streaming store, TH=NT
    }
  } else {
    v4f z = {0.0f, 0.0f, 0.0f, 0.0f};
#pragma unroll
    for (int k = 0; k < 4; ++k)
      __builtin_nontemporal_store(z, &op[lane + k * 32]);
  }
}

// ---------------------------------------------------------------------------
extern "C" void kernel_launch(void* const* d_in, const int* in_sizes, int n_in,
                              void* d_out, int out_size, void* d_ws, size_t ws_size,
                              hipStream_t stream) {
  (void)in_sizes; (void)n_in; (void)out_size; (void)ws_size;
  const float* x   = (const float*)d_in[0];   // [B, T, D] f32
  const float* dur = (const float*)d_in[1];   // [B, T]    f32
  float* out = (float*)d_out;                 // [B, T_OUT, D] f32
  int* ends = (int*)d_ws;                     // [B, T] i32 scratch (32 KB)

  lr_scan_kernel<<<BB, TT, 0, stream>>>(dur, ends);
  lr_expand_kernel<<<BB * (TOUT / 8), 256, 0, stream>>>(x, ends, out);
}